// ExcitationShaper_89120571392189
// MI455X (gfx1250) — compile-verified
//
#include <hip/hip_runtime.h>
#include <hip/hip_bf16.h>

// Problem constants (match reference: B=32, T=65536)
#define B_   32
#define T_   65536
#define CS   1024            // samples per chunk (one wave32 per chunk)
#define LS   32              // samples per lane  (CS/32)
#define NCH  (T_/CS)         // 64 chunks per row

typedef __attribute__((ext_vector_type(2))) float v2f;
typedef __attribute__((ext_vector_type(8))) float v8f;

// log-space mapping constants
#define LOG_D 2.99573227355399099f   // ln(2.0/0.1)  = ln 20
#define LOG_W 5.99146454710798199f   // ln(pi / (2*pi*20/16000)) = ln 400
#define LOG_Q 1.38629436111989062f   // ln(2.0/0.5)  = ln 4
#define MIN_D 0.1f
#define MIN_W 0.007853981633974483f  // 2*pi*20/16000
#define MIN_Q 0.5f

__device__ __forceinline__ float sigf(float x) { return 1.0f / (1.0f + expf(-x)); }

// ---------------------------------------------------------------------------
// K1: per-row segmented-boundary scan (deterministic, no atomics).
//   seg[r][t]  = inclusive cumsum of onsets  (segment id per element)
//   bnd[r][s]  = exclusive prefix sum of params at the start of segment s
//   pos[r][s]  = time index where segment s starts;  pos[r][nseg] = T
// Segment mean later = (bnd[s+1]-bnd[s]) / max(pos[s+1]-pos[s],1).
// ---------------------------------------------------------------------------
__global__ __launch_bounds__(1024) void k_segscan(
    const int* __restrict__ onsets, const float4* __restrict__ params,
    int* __restrict__ seg, float4* __restrict__ bnd, int* __restrict__ pos)
{
  const int r    = blockIdx.x;
  const int tid  = threadIdx.x;
  const int lane = tid & 31;
  const int wv   = tid >> 5;

  __shared__ int   sO[32];
  __shared__ float s0[32], s1[32], s2[32], s3[32];
  __shared__ int   cO;
  __shared__ float c0, c1, c2, c3;

  float4* bndr = bnd + (size_t)r * (T_ + 2);
  int*    posr = pos + (size_t)r * (T_ + 2);

  if (tid == 0) {
    cO = 0; c0 = 0.f; c1 = 0.f; c2 = 0.f; c3 = 0.f;
    bndr[0] = make_float4(0.f, 0.f, 0.f, 0.f);
    posr[0] = 0;
  }
  __syncthreads();

  for (int tile = 0; tile < T_ / 1024; ++tile) {
    const int    t   = tile * 1024 + tid;
    const size_t idx = (size_t)r * T_ + t;

    int    o  = onsets[idx];
    float4 pv = params[idx];
    float  p0 = pv.x, p1 = pv.y, p2 = pv.z, p3 = pv.w;

    // wave-level inclusive scan (onset count + 4 param channels)
    int   oi = o;
    float q0 = p0, q1 = p1, q2 = p2, q3 = p3;
    #pragma unroll
    for (int d = 1; d < 32; d <<= 1) {
      int   ou = __shfl_up(oi, d, 32);
      float u0 = __shfl_up(q0, d, 32), u1 = __shfl_up(q1, d, 32);
      float u2 = __shfl_up(q2, d, 32), u3 = __shfl_up(q3, d, 32);
      if (lane >= d) { oi += ou; q0 += u0; q1 += u1; q2 += u2; q3 += u3; }
    }
    if (lane == 31) { sO[wv] = oi; s0[wv] = q0; s1[wv] = q1; s2[wv] = q2; s3[wv] = q3; }
    __syncthreads();

    // wave 0 scans the 32 wave partials (inclusive)
    if (wv == 0) {
      int   vo = sO[lane];
      float v0 = s0[lane], v1 = s1[lane], v2 = s2[lane], v3 = s3[lane];
      #pragma unroll
      for (int d = 1; d < 32; d <<= 1) {
        int   ou = __shfl_up(vo, d, 32);
        float u0 = __shfl_up(v0, d, 32), u1 = __shfl_up(v1, d, 32);
        float u2 = __shfl_up(v2, d, 32), u3 = __shfl_up(v3, d, 32);
        if (lane >= d) { vo += ou; v0 += u0; v1 += u1; v2 += u2; v3 += u3; }
      }
      sO[lane] = vo; s0[lane] = v0; s1[lane] = v1; s2[lane] = v2; s3[lane] = v3;
    }
    __syncthreads();

    const int   offO = wv ? sO[wv - 1] : 0;
    const float f0o  = wv ? s0[wv - 1] : 0.f, f1o = wv ? s1[wv - 1] : 0.f;
    const float f2o  = wv ? s2[wv - 1] : 0.f, f3o = wv ? s3[wv - 1] : 0.f;

    const int segIncl = cO + offO + oi;            // inclusive onset cumsum
    const float e0 = c0 + f0o + q0 - p0;           // exclusive param prefix
    const float e1 = c1 + f1o + q1 - p1;
    const float e2 = c2 + f2o + q2 - p2;
    const float e3 = c3 + f3o + q3 - p3;

    seg[idx] = segIncl;
    if (o) {                                       // this element opens segment segIncl
      bndr[segIncl] = make_float4(e0, e1, e2, e3);
      posr[segIncl] = t;
    }
    __syncthreads();
    if (tid == 0) { cO += sO[31]; c0 += s0[31]; c1 += s1[31]; c2 += s2[31]; c3 += s3[31]; }
    __syncthreads();
  }
  if (tid == 0) {                                  // close the last segment
    bndr[cO + 1] = make_float4(c0, c1, c2, c3);
    posr[cO + 1] = T_;
  }
}

// ---------------------------------------------------------------------------
// K2: elementwise — segment mean, sigmoids, log-space maps, input scaling,
//     and per-sample biquad coefficients (b0, b1, a1, a2) with b2 == b0.
// ---------------------------------------------------------------------------
__global__ __launch_bounds__(256) void k_coef(
    const float* __restrict__ input, const int* __restrict__ seg,
    const float4* __restrict__ bnd, const int* __restrict__ pos,
    float* __restrict__ xs, float* __restrict__ mus, float4* __restrict__ cf)
{
  const size_t idx = (size_t)blockIdx.x * blockDim.x + threadIdx.x;
  const int r = (int)(idx >> 16);
  const int s = seg[idx];

  const float4* br = bnd + (size_t)r * (T_ + 2);
  const int*    pr = pos + (size_t)r * (T_ + 2);
  const float4  lo = br[s], hi = br[s + 1];
  int cnt = pr[s + 1] - pr[s];
  const float inv = 1.0f / (float)(cnt < 1 ? 1 : cnt);

  const float a0v = sigf((hi.x - lo.x) * inv);   // -> distance
  const float a1v = sigf((hi.y - lo.y) * inv);   // -> w_mod
  const float a2v = sigf((hi.z - lo.z) * inv);   // -> q_mod
  const float a3v = sigf((hi.w - lo.w) * inv);   // -> mu

  const float distance = MIN_D * expf(a0v * LOG_D);
  const float w        = MIN_W * expf(a1v * LOG_W);
  const float q        = MIN_Q * expf(a2v * LOG_Q);

  mus[idx] = a3v;
  xs[idx]  = input[idx] * distance;

  const float cw = cosf(w), sw = sinf(w);
  const float al = sw / (2.0f * q);
  const float ia = 1.0f / (1.0f + al);
  const float b1 = (1.0f - cw) * ia;
  const float b0 = 0.5f * b1;
  const float A1 = -2.0f * cw * ia;
  const float A2 = (1.0f - al) * ia;
  cf[idx] = make_float4(b0, b1, A1, A2);
}

// ---------------------------------------------------------------------------
// K3: sparse FIR comb (data-dependent gather into the scaled signal xs).
// ---------------------------------------------------------------------------
__global__ __launch_bounds__(256) void k_comb(
    const float* __restrict__ f0, const float* __restrict__ mus,
    const float* __restrict__ xs, float* __restrict__ xc)
{
  const size_t idx = (size_t)blockIdx.x * blockDim.x + threadIdx.x;
  const int r = (int)(idx >> 16);
  const int t = (int)(idx & (T_ - 1));

  const float p    = f0[idx] * mus[idx];
  const float z    = floorf(p);
  const float alfa = p - z;
  const int   zi   = (int)z;
  const int   i1   = t - (zi + 1);
  const int   i2   = t - (zi + 2);
  const size_t rb  = (size_t)r << 16;

  const float g1 = (i1 >= 0) ? xs[rb + i1] : 0.0f;
  const float g2 = (i2 >= 0) ? xs[rb + i2] : 0.0f;
  xc[idx] = xs[idx] - (1.0f - alfa) * g1 - alfa * g2;
}

// ---------------------------------------------------------------------------
// Affine 2-state operator: s' = M s + v   (companion form of the biquad).
// ---------------------------------------------------------------------------
struct Aff { float m00, m01, m10, m11, v0, v1; };

__device__ __forceinline__ Aff compose(const Aff& a /*newer*/, const Aff& b /*older*/) {
  Aff r;
  r.m00 = a.m00 * b.m00 + a.m01 * b.m10;
  r.m01 = a.m00 * b.m01 + a.m01 * b.m11;
  r.m10 = a.m10 * b.m00 + a.m11 * b.m10;
  r.m11 = a.m10 * b.m01 + a.m11 * b.m11;
  r.v0  = a.m00 * b.v0  + a.m01 * b.v1 + a.v0;
  r.v1  = a.m10 * b.v0  + a.m11 * b.v1 + a.v1;
  return r;
}

#define SHFL_AFF(dst, src, d)                                            \
  do {                                                                   \
    dst.m00 = __shfl_up(src.m00, d, 32); dst.m01 = __shfl_up(src.m01, d, 32); \
    dst.m10 = __shfl_up(src.m10, d, 32); dst.m11 = __shfl_up(src.m11, d, 32); \
    dst.v0  = __shfl_up(src.v0,  d, 32); dst.v1  = __shfl_up(src.v1,  d, 32); \
  } while (0)

// ---------------------------------------------------------------------------
// K4: wave-per-chunk. Each lane serially composes 32 steps, then a 5-level
//     wave shuffle-scan produces per-lane exclusive prefixes (stored) and the
//     chunk's total transfer operator (stored).
// ---------------------------------------------------------------------------
__global__ __launch_bounds__(256) void k_chunk(
    const float* __restrict__ xc, const float4* __restrict__ cf,
    float4* __restrict__ chunkH, float4* __restrict__ lanePre)
{
  const int gw   = (int)((blockIdx.x * blockDim.x + threadIdx.x) >> 5);
  const int lane = threadIdx.x & 31;
  const int r = gw / NCH, c = gw % NCH;
  const int t0 = c * CS + lane * LS;
  const size_t base = (size_t)r * T_ + t0;

  float x1 = 0.f, x2 = 0.f;
  if (t0 > 0) { x1 = xc[base - 1]; x2 = xc[base - 2]; }

  Aff A = {1.f, 0.f, 0.f, 1.f, 0.f, 0.f};
  #pragma unroll 4
  for (int i = 0; i < LS; ++i) {
    __builtin_prefetch((const void*)(xc + base + i + 8), 0, 1);   // global_prefetch_b8
    const float4 k  = cf[base + i];
    const float  xn = xc[base + i];
    const float  w  = k.x * (xn + x2) + k.y * x1;                 // b2 == b0
    const float  a1 = k.z, a2 = k.w;
    Aff n;
    n.m00 = -a1 * A.m00 - a2 * A.m10;
    n.m01 = -a1 * A.m01 - a2 * A.m11;
    n.m10 =  A.m00;  n.m11 = A.m01;
    n.v0  =  w - a1 * A.v0 - a2 * A.v1;
    n.v1  =  A.v0;
    A = n;  x2 = x1;  x1 = xn;
  }

  // wave-level inclusive scan of affine operators (noncommutative)
  #pragma unroll
  for (int d = 1; d < 32; d <<= 1) {
    Aff o; SHFL_AFF(o, A, d);
    if (lane >= d) A = compose(A, o);
  }
  // exclusive prefix for replay
  Aff E; SHFL_AFF(E, A, 1);
  if (lane == 0) { E.m00 = 1.f; E.m01 = 0.f; E.m10 = 0.f; E.m11 = 1.f; E.v0 = 0.f; E.v1 = 0.f; }

  float4* lp = lanePre + ((size_t)gw * 32 + lane) * 2;
  lp[0] = make_float4(E.m00, E.m01, E.m10, E.m11);
  lp[1] = make_float4(E.v0, E.v1, 0.f, 0.f);
  if (lane == 31) {
    float4* ch = chunkH + (size_t)gw * 2;
    ch[0] = make_float4(A.m00, A.m01, A.m10, A.m11);
    ch[1] = make_float4(A.v0, A.v1, 0.f, 0.f);
  }
}

// ---------------------------------------------------------------------------
// Homogeneous 3x3 element accessor for an operator stored as
// [m00, m01, m10, m11, v0, v1] -> rows { m00 m01 v0 ; m10 m11 v1 ; 0 0 1 }.
// ---------------------------------------------------------------------------
__device__ __forceinline__ float hEl(const float* h, int i, int j) {
  if (i == 0) return (j == 0) ? h[0] : ((j == 1) ? h[1] : h[4]);
  if (i == 1) return (j == 0) ? h[2] : ((j == 1) ? h[3] : h[5]);
  return (j == 2) ? 1.f : 0.f;
}

// ---------------------------------------------------------------------------
// K5: cross-chunk scan with f32 WMMA on the live path.
// One wave per row. Level 1: compose the 32 adjacent chunk-operator pairs
// with V_WMMA_F32_16X16X4_F32 — A tile stacks 5 "newer" 3x3 homogeneous
// operators along M, B tile stacks the 5 matching "older" operators along N;
// diagonal 3x3 blocks of D are the pair products (K=3 fits in K=4, full f32).
// Level 2: lane 0 serially scans the 32 pair operators (depth 32 instead of
// 64) to get even-chunk incoming states; all lanes then fix up odd chunks.
// ---------------------------------------------------------------------------
__global__ __launch_bounds__(32) void k_cross_wmma(
    const float4* __restrict__ chunkH, float2* __restrict__ sIn)
{
  const int r    = blockIdx.x;
  const int lane = threadIdx.x;       // one full wave32, EXEC all ones at wmma
  const int half = lane >> 4;
  const int l15  = lane & 15;

  __shared__ float  H[NCH][8];        // row's chunk operators
  __shared__ float  P[NCH / 2][8];    // pair-composed operators
  __shared__ float2 sE[NCH / 2];      // even-chunk incoming states

  for (int k = lane; k < NCH; k += 32) {
    const float4 hm = chunkH[(size_t)(r * NCH + k) * 2 + 0];
    const float4 hv = chunkH[(size_t)(r * NCH + k) * 2 + 1];
    H[k][0] = hm.x; H[k][1] = hm.y; H[k][2] = hm.z; H[k][3] = hm.w;
    H[k][4] = hv.x; H[k][5] = hv.y; H[k][6] = 0.f; H[k][7] = 0.f;
  }
  __syncthreads();

  // 32 pairs, 5 per WMMA issue -> 7 issues (last one holds 2 blocks)
  for (int g = 0; g < 7; ++g) {
    const int np = (g == 6) ? 2 : 5;

    // A-tile (16x4, ISA layout: a[0]=K0|K2, a[1]=K1|K3 per half-wave):
    v2f a = {0.f, 0.f};
    {
      const int bi = l15 / 3, mr = l15 % 3;
      if (bi < np && l15 < 15) {
        const float* h  = H[(g * 5 + bi) * 2 + 1];    // newer operator H_{2k+1}
        const int    k0 = half ? 2 : 0;
        a.x = hEl(h, mr, k0);
        a.y = (k0 + 1 < 3) ? hEl(h, mr, k0 + 1) : 0.f;  // K=3 column is zero
      }
    }
    // B-tile (4x16, row-striped: b[0]=K0|K2 row, b[1]=K1|K3 row):
    v2f b = {0.f, 0.f};
    {
      const int bj = l15 / 3, nc = l15 % 3;
      if (bj < np && l15 < 15) {
        const float* h  = H[(g * 5 + bj) * 2];        // older operator H_{2k}
        const int    k0 = half ? 2 : 0;
        b.x = hEl(h, k0, nc);
        b.y = (k0 + 1 < 3) ? hEl(h, k0 + 1, nc) : 0.f;  // K=3 row is zero
      }
    }

    v8f d = {};
    d = __builtin_amdgcn_wmma_f32_16x16x4_f32(
        /*neg_a=*/false, a, /*neg_b=*/false, b,
        /*c_mod=*/(short)0, d, /*reuse_a=*/false, /*reuse_b=*/false);

    // scatter diagonal 3x3 blocks back to LDS as pair operators
    #pragma unroll
    for (int v = 0; v < 8; ++v) {
      const int M = half ? (8 + v) : v;
      const int N = l15;
      const int bi = M / 3, bj = N / 3;
      if (bi == bj && bi < np && M < 15 && N < 15) {
        const int i = M - 3 * bi, j = N - 3 * bj;
        if (i < 2) {                                  // row 2 is [0 0 1]
          const int slot = (j == 2) ? (4 + i) : (i * 2 + j);
          P[g * 5 + bi][slot] = d[v];
        }
      }
    }
  }
  __syncthreads();

  // serial scan of the 32 pair operators (even-chunk incoming states)
  if (lane == 0) {
    float s0 = 0.f, s1 = 0.f;
    for (int k = 0; k < NCH / 2; ++k) {
      sE[k] = make_float2(s0, s1);
      const float* p = P[k];
      const float n0 = p[0] * s0 + p[1] * s1 + p[4];
      const float n1 = p[2] * s0 + p[3] * s1 + p[5];
      s0 = n0; s1 = n1;
    }
  }
  __syncthreads();

  // lane k emits states for chunks 2k (even) and 2k+1 (odd fixup via H_{2k})
  {
    const int    k  = lane;
    const float2 se = sE[k];
    const float* h  = H[2 * k];
    const float  o0 = h[0] * se.x + h[1] * se.y + h[4];
    const float  o1 = h[2] * se.x + h[3] * se.y + h[5];
    sIn[r * NCH + 2 * k]     = se;
    sIn[r * NCH + 2 * k + 1] = make_float2(o0, o1);
  }
}

// ---------------------------------------------------------------------------
// K6: replay. Lane state = exclusive-prefix operator applied to the chunk's
//     incoming state, then 32 serial biquad steps emitting y.
// ---------------------------------------------------------------------------
__global__ __launch_bounds__(256) void k_replay(
    const float* __restrict__ xc, const float4* __restrict__ cf,
    const float4* __restrict__ lanePre, const float2* __restrict__ sIn,
    float* __restrict__ out)
{
  const int gw   = (int)((blockIdx.x * blockDim.x + threadIdx.x) >> 5);
  const int lane = threadIdx.x & 31;
  const int r = gw / NCH, c = gw % NCH;
  const int t0 = c * CS + lane * LS;
  const size_t base = (size_t)r * T_ + t0;

  const float2 s = sIn[r * NCH + c];
  const float4* lp = lanePre + ((size_t)gw * 32 + lane) * 2;
  const float4 em = lp[0], ev = lp[1];
  float y1 = em.x * s.x + em.y * s.y + ev.x;   // y[t0-1]
  float y2 = em.z * s.x + em.w * s.y + ev.y;   // y[t0-2]

  float x1 = 0.f, x2 = 0.f;
  if (t0 > 0) { x1 = xc[base - 1]; x2 = xc[base - 2]; }

  #pragma unroll 4
  for (int i = 0; i < LS; ++i) {
    __builtin_prefetch((const void*)(cf + base + i + 8), 0, 1);
    const float4 k  = cf[base + i];
    const float  xn = xc[base + i];
    const float  w  = k.x * (xn + x2) + k.y * x1;
    const float  y  = w - k.z * y1 - k.w * y2;
    out[base + i] = y;
    y2 = y1; y1 = y; x2 = x1; x1 = xn;
  }
}

// ---------------------------------------------------------------------------
extern "C" void kernel_launch(void* const* d_in, const int* in_sizes, int n_in,
                              void* d_out, int out_size, void* d_ws, size_t ws_size,
                              hipStream_t stream) {
  const float*  f0     = (const float*)d_in[0];
  const float*  input  = (const float*)d_in[1];
  const float4* params = (const float4*)d_in[2];
  const int*    onsets = (const int*)d_in[3];
  float*        out    = (float*)d_out;

  char*  w   = (char*)d_ws;
  size_t off = 0;
  auto alloc = [&](size_t bytes) -> void* {
    off = (off + 255) & ~(size_t)255;
    void* p = (void*)(w + off);
    off += bytes;
    return p;
  };
  int*    seg     = (int*)   alloc((size_t)B_ * T_ * 4);
  float4* bnd     = (float4*)alloc((size_t)B_ * (T_ + 2) * 16);
  int*    pos     = (int*)   alloc((size_t)B_ * (T_ + 2) * 4);
  float*  xs      = (float*) alloc((size_t)B_ * T_ * 4);
  float*  mus     = (float*) alloc((size_t)B_ * T_ * 4);
  float*  xc      = (float*) alloc((size_t)B_ * T_ * 4);
  float4* cf      = (float4*)alloc((size_t)B_ * T_ * 16);
  float4* chunkH  = (float4*)alloc((size_t)B_ * NCH * 32);
  float4* lanePre = (float4*)alloc((size_t)B_ * NCH * 32 * 32);
  float2* sIn     = (float2*)alloc((size_t)B_ * NCH * 8);
  (void)in_sizes; (void)n_in; (void)out_size; (void)ws_size;

  const int N = B_ * T_;
  k_segscan   <<<B_, 1024, 0, stream>>>(onsets, params, seg, bnd, pos);
  k_coef      <<<N / 256, 256, 0, stream>>>(input, seg, bnd, pos, xs, mus, cf);
  k_comb      <<<N / 256, 256, 0, stream>>>(f0, mus, xs, xc);
  k_chunk     <<<(B_ * NCH) / 8, 256, 0, stream>>>(xc, cf, chunkH, lanePre);
  k_cross_wmma<<<B_, 32, 0, stream>>>(chunkH, sIn);
  k_replay    <<<(B_ * NCH) / 8, 256, 0, stream>>>(xc, cf, lanePre, sIn, out);
}